// GCNDepth_80676665688567
// MI455X (gfx1250) — compile-verified
//
#include <hip/hip_runtime.h>
#include <hip/hip_bf16.h>

#define N_NODES 50000
#define N_EDGES 800000
#define IN_C 128
#define HID_C 64
#define OUT_C 40
#define BN_EPS 1e-5f

typedef __attribute__((ext_vector_type(2))) float v2f;
typedef __attribute__((ext_vector_type(8))) float v8f;

// ---------------- degree / normalization ----------------

__global__ void deg_init_kernel(float* __restrict__ deg, int n) {
    int i = blockIdx.x * blockDim.x + threadIdx.x;
    if (i < n) deg[i] = 1.0f;  // self-loop contribution
}

__global__ void deg_accum_kernel(const int* __restrict__ dst, float* deg, int ne) {
    int i = blockIdx.x * blockDim.x + threadIdx.x;
    if (i < ne) atomicAdd(&deg[dst[i]], 1.0f);
}

__global__ void deg_rsqrt_kernel(float* __restrict__ deg, int n) {
    int i = blockIdx.x * blockDim.x + threadIdx.x;
    if (i < n) deg[i] = rsqrtf(deg[i]);  // deg >= 1 always (self loops)
}

// ---- WMMA f32 GEMM + fused self-loop/bias epilogue ----
// H[n,COUT] = X[n,K] @ W[K,COUT];  AGG[n,COUT] = bias + H * dinv[row]^2
// One wave owns a 16-row slab; V_WMMA_F32_16X16X4_F32 over K in steps of 4.
// A (16x4): lanes 0-15 = M rows, {k0,k0+1}; lanes 16-31 = {k0+2,k0+3}.
// B (4x16): lanes 0-15 = N cols, rows {k0,k0+1}; lanes 16-31 = rows {k0+2,k0+3}.
// C/D (16x16): VGPR r = row r (lanes 0-15) / row r+8 (lanes 16-31), col = lane%16.
// Out-of-range rows/cols are CLAMPED (not masked): their results are garbage but
// never stored, and WMMA rows/cols are independent, so valid outputs are exact.

template <int K, int COUT, int NT>
__global__ __launch_bounds__(128) void gemm_wmma_kernel(
    const float* __restrict__ X, const float* __restrict__ W,
    const float* __restrict__ bias, const float* __restrict__ dinv,
    float* __restrict__ H, float* __restrict__ AGG, int nrows) {
    const int wave    = threadIdx.x >> 5;                  // 0..3
    const int lane    = threadIdx.x & 31;
    const int l15     = lane & 15;
    const int koff    = (lane >> 4) << 1;                  // 0 or 2
    const int rowBase = (blockIdx.x * 4 + wave) * 16;
    const int mrow    = min(rowBase + l15, nrows - 1);     // clamp: no exec masking

    int  colRaw[NT];
    int  col[NT];
    float biasv[NT];
#pragma unroll
    for (int t = 0; t < NT; ++t) {
        colRaw[t] = t * 16 + l15;
        col[t]    = ((COUT & 15) == 0) ? colRaw[t] : min(colRaw[t], COUT - 1);
        biasv[t]  = bias[col[t]];
    }

    v8f acc[NT];
#pragma unroll
    for (int t = 0; t < NT; ++t)
#pragma unroll
        for (int r = 0; r < 8; ++r) acc[t][r] = 0.0f;

    const float* Arow = X + (size_t)mrow * K;
    for (int k = 0; k < K; k += 4) {
        v2f a = *(const v2f*)(Arow + k + koff);            // single b64 load
#pragma unroll
        for (int t = 0; t < NT; ++t) {
            v2f b;
            b.x = W[(k + koff) * COUT + col[t]];
            b.y = W[(k + koff + 1) * COUT + col[t]];
            acc[t] = __builtin_amdgcn_wmma_f32_16x16x4_f32(
                false, a, false, b, (short)0, acc[t], false, false);
        }
    }

    const int rowHalf = (lane >> 4) << 3;  // 0 or 8
#pragma unroll
    for (int r = 0; r < 8; ++r) {
        const int row = rowBase + rowHalf + r;
        if (row < nrows) {
            const float di  = dinv[row];
            const float di2 = di * di;
#pragma unroll
            for (int t = 0; t < NT; ++t) {
                if (((COUT & 15) == 0) || (colRaw[t] < COUT)) {
                    const float v = acc[t][r];
                    H[row * COUT + colRaw[t]]   = v;
                    AGG[row * COUT + colRaw[t]] = fmaf(v, di2, biasv[t]);
                }
            }
        }
    }
}

// ---------------- edge scatter: agg[dst] += h[src] * dinv[src]*dinv[dst] ----------------
// float4 per thread: 16 (C=64) or 10 (C=40) threads per edge; coalesced b128 h reads,
// 4x fewer redundant src/dst/dinv loads. Atomic f32 adds are the irreducible cost.

template <int C>
__global__ void scatter_kernel(const float4* __restrict__ h4, const int* __restrict__ src,
                               const int* __restrict__ dst, const float* __restrict__ dinv,
                               float* agg, int ne) {
    constexpr int CG = C / 4;
    int g = blockIdx.x * blockDim.x + threadIdx.x;  // ne*CG <= 12.8M, fits int32
    if (g >= ne * CG) return;
    int e  = g / CG;
    int cg = g % CG;
    int s = src[e];
    int d = dst[e];
    float norm = dinv[s] * dinv[d];
    float4 v = h4[s * CG + cg];
    float* base = agg + d * C + cg * 4;
    atomicAdd(base + 0, v.x * norm);
    atomicAdd(base + 1, v.y * norm);
    atomicAdd(base + 2, v.z * norm);
    atomicAdd(base + 3, v.w * norm);
}

// ---------------- eval-mode BN + ReLU (C = HID_C = 64), float4 ----------------

__global__ void bn_relu_kernel(const float4* __restrict__ in, float4* __restrict__ out,
                               const float4* __restrict__ gamma, const float4* __restrict__ beta,
                               const float4* __restrict__ mean, const float4* __restrict__ var,
                               int n) {
    constexpr int CG = HID_C / 4;  // 16
    int i = blockIdx.x * blockDim.x + threadIdx.x;
    if (i >= n * CG) return;
    int cg = i & (CG - 1);
    float4 x = in[i], m = mean[cg], v = var[cg], g = gamma[cg], b = beta[cg];
    float4 o;
    o.x = fmaxf((x.x - m.x) * rsqrtf(v.x + BN_EPS) * g.x + b.x, 0.0f);
    o.y = fmaxf((x.y - m.y) * rsqrtf(v.y + BN_EPS) * g.y + b.y, 0.0f);
    o.z = fmaxf((x.z - m.z) * rsqrtf(v.z + BN_EPS) * g.z + b.z, 0.0f);
    o.w = fmaxf((x.w - m.w) * rsqrtf(v.w + BN_EPS) * g.w + b.w, 0.0f);
    out[i] = o;
}

// ---------------- launch ----------------

static inline int cdiv_ll(long long a, long long b) { return (int)((a + b - 1) / b); }

extern "C" void kernel_launch(void* const* d_in, const int* in_sizes, int n_in,
                              void* d_out, int out_size, void* d_ws, size_t ws_size,
                              hipStream_t stream) {
    (void)in_sizes; (void)n_in; (void)out_size; (void)ws_size;

    const float* x   = (const float*)d_in[0];
    const int*   ei  = (const int*)d_in[1];   // edge_index [2, E] (int32: JAX x64 disabled)
    const float* W0  = (const float*)d_in[2];
    const float* b0  = (const float*)d_in[3];
    const float* g0  = (const float*)d_in[4];
    const float* be0 = (const float*)d_in[5];
    const float* m0  = (const float*)d_in[6];
    const float* v0  = (const float*)d_in[7];
    const float* W1  = (const float*)d_in[8];
    const float* b1  = (const float*)d_in[9];
    const float* g1  = (const float*)d_in[10];
    const float* be1 = (const float*)d_in[11];
    const float* m1  = (const float*)d_in[12];
    const float* v1  = (const float*)d_in[13];
    const float* W2  = (const float*)d_in[14];
    const float* b2  = (const float*)d_in[15];

    const int* src = ei;
    const int* dst = ei + N_EDGES;

    float* ws   = (float*)d_ws;
    float* dinv = ws;                                   // N
    float* h    = dinv + N_NODES;                       // N*64
    float* agg  = h + (size_t)N_NODES * HID_C;          // N*64
    float* act  = agg + (size_t)N_NODES * HID_C;        // N*64
    float* out  = (float*)d_out;                        // N*40

    const int B = 256;
    const int nNodeB = cdiv_ll(N_NODES, B);
    const int nEdgeB = cdiv_ll(N_EDGES, B);
    const int nBN    = cdiv_ll((long long)N_NODES * (HID_C / 4), B);
    const int nSc64  = cdiv_ll((long long)N_EDGES * (HID_C / 4), B);
    const int nSc40  = cdiv_ll((long long)N_EDGES * (OUT_C / 4), B);
    const int gemmB  = (N_NODES + 63) / 64;             // 4 waves x 16 rows per block

    // normalization (shared by all layers)
    deg_init_kernel<<<nNodeB, B, 0, stream>>>(dinv, N_NODES);
    deg_accum_kernel<<<nEdgeB, B, 0, stream>>>(dst, dinv, N_EDGES);
    deg_rsqrt_kernel<<<nNodeB, B, 0, stream>>>(dinv, N_NODES);

    // layer 0: 128 -> 64
    gemm_wmma_kernel<IN_C, HID_C, 4><<<gemmB, 128, 0, stream>>>(x, W0, b0, dinv, h, agg, N_NODES);
    scatter_kernel<HID_C><<<nSc64, B, 0, stream>>>((const float4*)h, src, dst, dinv, agg, N_EDGES);
    bn_relu_kernel<<<nBN, B, 0, stream>>>((const float4*)agg, (float4*)act, (const float4*)g0,
                                          (const float4*)be0, (const float4*)m0, (const float4*)v0, N_NODES);

    // layer 1: 64 -> 64
    gemm_wmma_kernel<HID_C, HID_C, 4><<<gemmB, 128, 0, stream>>>(act, W1, b1, dinv, h, agg, N_NODES);
    scatter_kernel<HID_C><<<nSc64, B, 0, stream>>>((const float4*)h, src, dst, dinv, agg, N_EDGES);
    bn_relu_kernel<<<nBN, B, 0, stream>>>((const float4*)agg, (float4*)act, (const float4*)g1,
                                          (const float4*)be1, (const float4*)m1, (const float4*)v1, N_NODES);

    // layer 2: 64 -> 40 (self-loop + bias epilogue writes straight into d_out)
    gemm_wmma_kernel<HID_C, OUT_C, 3><<<gemmB, 128, 0, stream>>>(act, W2, b2, dinv, h, out, N_NODES);
    scatter_kernel<OUT_C><<<nSc40, B, 0, stream>>>((const float4*)h, src, dst, dinv, out, N_EDGES);
}